// QLSTM_40450001993934
// MI455X (gfx1250) — compile-verified
//
#include <hip/hip_runtime.h>

// QLSTM scan: B=8192 independent rows, T=4096 serial steps each.
// One thread per row; state (c, h1..h3, a) in registers. Latency-bound on the
// serial chain -> gfx1250 v_tanh_f32 activations; input streamed through LDS
// with GLOBAL_LOAD_ASYNC_TO_LDS_B128 double-buffering (ASYNCcnt-tracked),
// falling back to float4 loads + global_prefetch_b8 if builtins are absent.

#define QL_THREADS 64   // small blocks -> 128 blocks spread across WGPs

#if __has_builtin(__builtin_amdgcn_global_load_async_to_lds_b128) && \
    __has_builtin(__builtin_amdgcn_s_wait_asynccnt)
#define QL_ASYNC 1
#else
#define QL_ASYNC 0
#endif

#if QL_ASYNC
typedef int ql_v4i __attribute__((ext_vector_type(4)));
typedef __attribute__((address_space(1))) ql_v4i* ql_gptr;   // global (AS1)
typedef __attribute__((address_space(3))) ql_v4i* ql_lptr;   // LDS (AS3)
#endif

__device__ __forceinline__ float fast_tanh(float x) {
#if defined(__AMDGCN__) && __has_builtin(__builtin_amdgcn_tanhf)
    return __builtin_amdgcn_tanhf(x);            // gfx1250 v_tanh_f32 (1 trans op)
#else
    float e = __expf(2.0f * x);                  // v_exp_f32 path
    return 1.0f - 2.0f * __builtin_amdgcn_rcpf(e + 1.0f);
#endif
}

__device__ __forceinline__ float fast_sigmoid(float x) {
    // sigmoid(x) = 0.5*tanh(0.5x) + 0.5 -> single transcendental on the chain
    return fmaf(0.5f, fast_tanh(0.5f * x), 0.5f);
}

struct QLW {
    float wf0, wf1, wf2, wf3, bf;
    float wi0, wi1, wi2, wi3, bi;
    float wg0, wg1, wg2, wg3, bg;
    float wo0, wo1, wo2, wo3, bo;
    float whS0, whS1, whS2, bh0, bh1, bh2;
    float wyS, by;
};

struct QLS { float c, h1, h2, h3, a; };

__device__ __forceinline__ void ql_step(QLS& s, float x, const QLW& w) {
    // x-dependent FMA is recurrence-independent -> overlaps previous step.
    float pf = fmaf(s.h1, w.wf1, fmaf(s.h2, w.wf2, fmaf(s.h3, w.wf3, fmaf(x, w.wf0, w.bf))));
    float pi = fmaf(s.h1, w.wi1, fmaf(s.h2, w.wi2, fmaf(s.h3, w.wi3, fmaf(x, w.wi0, w.bi))));
    float pg = fmaf(s.h1, w.wg1, fmaf(s.h2, w.wg2, fmaf(s.h3, w.wg3, fmaf(x, w.wg0, w.bg))));
    float po = fmaf(s.h1, w.wo1, fmaf(s.h2, w.wo2, fmaf(s.h3, w.wo3, fmaf(x, w.wo0, w.bo))));
    float f  = fast_sigmoid(pf);
    float i_ = fast_sigmoid(pi);
    float g  = fast_tanh(pg);
    float o  = fast_sigmoid(po);
    s.c  = fmaf(f, s.c, i_ * g);        // ct components identical -> scalar cell
    s.a  = o * fast_tanh(s.c);          // scalar "act"
    s.h1 = fmaf(s.a, w.whS0, w.bh0);    // ht via precomputed Wh row-sums
    s.h2 = fmaf(s.a, w.whS1, w.bh1);
    s.h3 = fmaf(s.a, w.whS2, w.bh2);
}

__global__ __launch_bounds__(QL_THREADS) void qlstm_scan_kernel(
    const float* __restrict__ xseq,                                   // [B,T,1]
    const float* __restrict__ Wf, const float* __restrict__ bf,
    const float* __restrict__ Wi, const float* __restrict__ bi,
    const float* __restrict__ Wg, const float* __restrict__ bg,
    const float* __restrict__ Wo, const float* __restrict__ bo,
    const float* __restrict__ Wy, const float* __restrict__ by,
    const float* __restrict__ Wh, const float* __restrict__ bh,
    float* __restrict__ yout,                                         // [B,1]
    int B, int T)
{
    const int b = blockIdx.x * blockDim.x + threadIdx.x;
    if (b >= B) return;

    // ---- wave-uniform weights -> SGPR broadcasts ----
    QLW w;
    w.wf0 = Wf[0]; w.wf1 = Wf[1]; w.wf2 = Wf[2]; w.wf3 = Wf[3]; w.bf = bf[0];
    w.wi0 = Wi[0]; w.wi1 = Wi[1]; w.wi2 = Wi[2]; w.wi3 = Wi[3]; w.bi = bi[0];
    w.wg0 = Wg[0]; w.wg1 = Wg[1]; w.wg2 = Wg[2]; w.wg3 = Wg[3]; w.bg = bg[0];
    w.wo0 = Wo[0]; w.wo1 = Wo[1]; w.wo2 = Wo[2]; w.wo3 = Wo[3]; w.bo = bo[0];
    w.whS0 = ((Wh[0] + Wh[1]) + (Wh[2] + Wh[3]));
    w.whS1 = ((Wh[4] + Wh[5]) + (Wh[6] + Wh[7]));
    w.whS2 = ((Wh[8] + Wh[9]) + (Wh[10] + Wh[11]));
    w.bh0 = bh[0]; w.bh1 = bh[1]; w.bh2 = bh[2];
    w.wyS = ((Wy[0] + Wy[1]) + (Wy[2] + Wy[3]));
    w.by  = by[0];

    QLS s = {0.0f, 0.0f, 0.0f, 0.0f, 0.0f};

    const float* xrow = xseq + (size_t)b * (size_t)T;
    const int nT4 = T >> 2;
    const bool vec = ((T & 3) == 0) && (((uintptr_t)xrow & 15u) == 0) && (nT4 > 0);

    if (vec) {
        const float4* xp = (const float4*)xrow;
#if QL_ASYNC
        // ---- gfx1250 async global->LDS double-buffered input stream ----
        // Each lane owns a private 16B slot per buffer: no barriers needed,
        // completion ordered via ASYNCcnt (async loads retire in order).
        __shared__ float4 stage[2][QL_THREADS];
        const int lt = threadIdx.x;
        __builtin_amdgcn_global_load_async_to_lds_b128(
            (ql_gptr)(xp + 0), (ql_lptr)(&stage[0][lt]), 0, 0);
        for (int q = 0; q < nT4; ++q) {
            const int cur = q & 1;
            if (q + 1 < nT4) {
                __builtin_amdgcn_global_load_async_to_lds_b128(
                    (ql_gptr)(xp + q + 1), (ql_lptr)(&stage[cur ^ 1][lt]), 0, 0);
                __builtin_amdgcn_s_wait_asynccnt(1);   // chunk q has landed
            } else {
                __builtin_amdgcn_s_wait_asynccnt(0);   // last chunk landed
            }
            float4 xv = stage[cur][lt];                // ds_load_b128
            ql_step(s, xv.x, w);
            ql_step(s, xv.y, w);
            ql_step(s, xv.z, w);
            ql_step(s, xv.w, w);
        }
#else
        // ---- fallback: register pipeline + gfx1250 global_prefetch_b8 ----
        for (int q = 0; q < nT4; ++q) {
            __builtin_prefetch(&xp[q + 16], 0, 1);     // ~256B / 64 steps ahead
            float4 xv = xp[q];
            ql_step(s, xv.x, w);
            ql_step(s, xv.y, w);
            ql_step(s, xv.z, w);
            ql_step(s, xv.w, w);
        }
#endif
        for (int t = nT4 << 2; t < T; ++t) ql_step(s, xrow[t], w);
    } else {
        for (int t = 0; t < T; ++t) ql_step(s, xrow[t], w);
    }

    yout[b] = fmaf(s.a, w.wyS, w.by);   // only the final yt is returned
}

extern "C" void kernel_launch(void* const* d_in, const int* in_sizes, int n_in,
                              void* d_out, int out_size, void* d_ws, size_t ws_size,
                              hipStream_t stream) {
    (void)n_in; (void)d_ws; (void)ws_size;

    const float* xseq = (const float*)d_in[0];
    const float* Wf = (const float*)d_in[1];  const float* bf = (const float*)d_in[2];
    const float* Wi = (const float*)d_in[3];  const float* bi = (const float*)d_in[4];
    const float* Wg = (const float*)d_in[5];  const float* bg = (const float*)d_in[6];
    const float* Wo = (const float*)d_in[7];  const float* bo = (const float*)d_in[8];
    const float* Wy = (const float*)d_in[9];  const float* by = (const float*)d_in[10];
    const float* Wh = (const float*)d_in[11]; const float* bh = (const float*)d_in[12];

    const int B = out_size;                       // output is [B, 1] float32
    const int T = (B > 0) ? (in_sizes[0] / B) : 0;

    const int threads = QL_THREADS;
    const int blocks  = (B + threads - 1) / threads;   // 128 blocks for B=8192
    qlstm_scan_kernel<<<blocks, threads, 0, stream>>>(
        xseq, Wf, bf, Wi, bi, Wg, bg, Wo, bo, Wy, by, Wh, bh,
        (float*)d_out, B, T);
}